// SpGraphAttentionLayerV0_1632087573323
// MI455X (gfx1250) — compile-verified
//
#include <hip/hip_runtime.h>
#include <hip/hip_bf16.h>
#include <math.h>

typedef __attribute__((ext_vector_type(2))) float v2f;
typedef __attribute__((ext_vector_type(8))) float v8f;

#define C_CURV   1.0f
#define SQRT_C   1.0f
#define ALPHA    0.2f
#define MIN_NORM 1e-15f
#define ATANH_CLIP (1.0f - 1e-7f)

__device__ __forceinline__ float artanh_f(float x) {
    x = fminf(fmaxf(x, -ATANH_CLIP), ATANH_CLIP);
    return 0.5f * (log1pf(x) - log1pf(-x));
}

// ---------------------------------------------------------------------------
// K0: zero scratch (h_prime, e_rowsum, ssq are laid out contiguously)
// ---------------------------------------------------------------------------
__global__ void zero_kernel(float* __restrict__ p, int n) {
    int i = blockIdx.x * blockDim.x + threadIdx.x;
    if (i < n) p[i] = 0.0f;
}

// ---------------------------------------------------------------------------
// K1: hyperbolic linear layer via V_WMMA_F32_16X16X4_F32.
// One wave computes a 16-node x 64-output tile: mx = x @ W^T (fp32 WMMA),
// then per-row norms via half-wave shuffles, then h = mobius scaling and
// h_e = logmap0(h), both written to scratch.
//
// A (16x4 f32): lane<16 holds {K=k0,k0+1} of row M=lane; lane>=16 holds
// {K=k0+2,k0+3} of row M=lane-16  -> contiguous v2f at x + row*64 + k0 + 2*half.
// B (4x16) mirrors A on W rows (o = o_tile*16 + (lane&15)).
// C/D (16x16 f32): VGPR r = row r (lanes 0-15) / row r+8 (lanes 16-31).
// ---------------------------------------------------------------------------
__global__ void hyp_linear_kernel(const float* __restrict__ x,
                                  const float* __restrict__ w,
                                  float* __restrict__ h,
                                  float* __restrict__ h_e,
                                  int N, int ntiles) {
    int wave = threadIdx.x >> 5;
    int lane = threadIdx.x & 31;
    int tile = blockIdx.x * (blockDim.x >> 5) + wave;
    if (tile >= ntiles) return;

    int base = tile * 16;
    int half = lane >> 4;    // 0 or 1
    int lr   = lane & 15;

    int arow = base + lr;
    if (arow > N - 1) arow = N - 1;
    const float* xrow = x + (size_t)arow * 64 + 2 * half;
    const float* wrow = w + (size_t)lr * 64 + 2 * half;

    v8f acc0 = {}, acc1 = {}, acc2 = {}, acc3 = {};
    float xn2p = 0.0f;

#pragma unroll
    for (int k = 0; k < 16; ++k) {
        int k0 = 4 * k;
        v2f a  = *(const v2f*)(xrow + k0);
        xn2p += a.x * a.x + a.y * a.y;
        v2f b0 = *(const v2f*)(wrow + k0);
        v2f b1 = *(const v2f*)(wrow + 16 * 64 + k0);
        v2f b2 = *(const v2f*)(wrow + 32 * 64 + k0);
        v2f b3 = *(const v2f*)(wrow + 48 * 64 + k0);
        acc0 = __builtin_amdgcn_wmma_f32_16x16x4_f32(false, a, false, b0, (short)0, acc0, false, false);
        acc1 = __builtin_amdgcn_wmma_f32_16x16x4_f32(false, a, false, b1, (short)0, acc1, false, false);
        acc2 = __builtin_amdgcn_wmma_f32_16x16x4_f32(false, a, false, b2, (short)0, acc2, false, false);
        acc3 = __builtin_amdgcn_wmma_f32_16x16x4_f32(false, a, false, b3, (short)0, acc3, false, false);
    }

    // ||x||^2 per node: lanes L and L+16 hold complementary K-halves of node base+(L&15)
    float xn2 = xn2p + __shfl_xor(xn2p, 16, 32);

#pragma unroll
    for (int r = 0; r < 8; ++r) {
        float m0 = acc0[r], m1 = acc1[r], m2 = acc2[r], m3 = acc3[r];
        // ||mx||^2 for node (base + r + 8*half): reduce over 16 lanes of this half
        float mp = m0 * m0 + m1 * m1 + m2 * m2 + m3 * m3;
        mp += __shfl_xor(mp, 8, 32);
        mp += __shfl_xor(mp, 4, 32);
        mp += __shfl_xor(mp, 2, 32);
        mp += __shfl_xor(mp, 1, 32);
        // fetch this node's ||x||^2 (held by lane r / r+8)
        float xn2n = __shfl(xn2, r + 8 * half, 32);

        float x_norm  = fmaxf(sqrtf(xn2n), MIN_NORM);
        float mx_norm = fmaxf(sqrtf(mp), MIN_NORM);
        float at = artanh_f(SQRT_C * x_norm);
        float sc = tanhf(mx_norm / x_norm * at) / (mx_norm * SQRT_C);   // h = sc * mx
        float h_norm = fmaxf(sc * mx_norm, MIN_NORM);                   // sc >= 0
        float le = artanh_f(SQRT_C * h_norm) / (SQRT_C * h_norm);       // logmap0 factor

        int node = base + r + 8 * half;
        if (node < N) {
            float* hr  = h   + (size_t)node * 64 + lr;
            float* her = h_e + (size_t)node * 64 + lr;
            hr[0]  = sc * m0;  hr[16]  = sc * m1;  hr[32]  = sc * m2;  hr[48]  = sc * m3;
            her[0] = le * sc * m0; her[16] = le * sc * m1;
            her[32] = le * sc * m2; her[48] = le * sc * m3;
        }
    }
}

// ---------------------------------------------------------------------------
// K2: per-edge Poincare sqdist -> edge_h = C/(C+dist). One wave per edge,
// float2 per lane, full-wave shuffle reductions. Gathers hit L2 (h = 12.8MB).
// ---------------------------------------------------------------------------
__global__ void edge_dist_kernel(const float* __restrict__ h,
                                 const int* __restrict__ erow,
                                 const int* __restrict__ ecol,
                                 float* __restrict__ edge_h,
                                 int E) {
    int wave = threadIdx.x >> 5;
    int lane = threadIdx.x & 31;
    int e = blockIdx.x * (blockDim.x >> 5) + wave;
    if (e >= E) return;

    int r = erow[e], c = ecol[e];
    float2 p1 = *((const float2*)(h + (size_t)r * 64) + lane);
    float2 p2 = *((const float2*)(h + (size_t)c * 64) + lane);

    float sx  = p1.x * p1.x + p1.y * p1.y;
    float sy  = p2.x * p2.x + p2.y * p2.y;
    float sxy = p1.x * p2.x + p1.y * p2.y;
#pragma unroll
    for (int o = 16; o >= 1; o >>= 1) {
        sx  += __shfl_xor(sx, o, 32);
        sy  += __shfl_xor(sy, o, 32);
        sxy += __shfl_xor(sxy, o, 32);
    }
    // mobius_add(-p1, p2, c): x=-p1 so x2=sx, y2=sy, xy=-sxy
    float xy = -sxy;
    float ca = 1.0f + 2.0f * C_CURV * xy + C_CURV * sy;
    float cb = 1.0f - C_CURV * sx;
    float nx = ca * (-p1.x) + cb * p2.x;
    float ny = ca * (-p1.y) + cb * p2.y;
    float nn = nx * nx + ny * ny;
#pragma unroll
    for (int o = 16; o >= 1; o >>= 1) nn += __shfl_xor(nn, o, 32);

    float denom  = fmaxf(1.0f + 2.0f * C_CURV * xy + C_CURV * C_CURV * sx * sy, MIN_NORM);
    float manorm = sqrtf(nn) / denom;
    float d = (2.0f / SQRT_C) * artanh_f(SQRT_C * manorm);
    float dist = d * d;
    if (lane == 0) edge_h[e] = C_CURV / (C_CURV + dist);
}

// ---------------------------------------------------------------------------
// K3: global sum of squares of edge_h (logmap0 over the edge axis norms the
// whole E-vector). Grid-stride + wave reduce + one atomic per wave.
// ---------------------------------------------------------------------------
__global__ void ssq_kernel(const float* __restrict__ eh, float* __restrict__ ssq, int E) {
    float s = 0.0f;
    for (int i = blockIdx.x * blockDim.x + threadIdx.x; i < E; i += gridDim.x * blockDim.x) {
        float v = eh[i];
        s += v * v;
    }
#pragma unroll
    for (int o = 16; o >= 1; o >>= 1) s += __shfl_xor(s, o, 32);
    if ((threadIdx.x & 31) == 0) atomicAdd(ssq, s);
}

// ---------------------------------------------------------------------------
// K4: edge_e = exp(-leakyrelu(edge_h * artanh(P)/P)); scatter-add into
// h_prime[row] (64 atomics/edge) and e_rowsum[row]. Atomics resolve in L2.
// ---------------------------------------------------------------------------
__global__ void scatter_kernel(const float* __restrict__ edge_h,
                               const float* __restrict__ h_e,
                               const int* __restrict__ erow,
                               const int* __restrict__ ecol,
                               const float* __restrict__ ssq,
                               float* __restrict__ h_prime,
                               float* __restrict__ e_rowsum,
                               int E) {
    int wave = threadIdx.x >> 5;
    int lane = threadIdx.x & 31;
    int e = blockIdx.x * (blockDim.x >> 5) + wave;
    if (e >= E) return;

    float pnorm = fmaxf(sqrtf(*ssq), MIN_NORM);
    float f = artanh_f(SQRT_C * pnorm) / (SQRT_C * pnorm);

    float v   = edge_h[e] * f;
    float lrl = (v >= 0.0f) ? v : ALPHA * v;
    float ee  = expf(-lrl);

    int r = erow[e], c = ecol[e];
    float2 he = *((const float2*)(h_e + (size_t)c * 64) + lane);
    float* dst = h_prime + (size_t)r * 64 + lane * 2;
    atomicAdd(dst,     ee * he.x);
    atomicAdd(dst + 1, ee * he.y);
    if (lane == 0) atomicAdd(e_rowsum + r, ee);
}

// ---------------------------------------------------------------------------
// K5: out = proj(expmap0(elu(h_prime / e_rowsum))). One wave per node.
// ---------------------------------------------------------------------------
__global__ void out_kernel(const float* __restrict__ h_prime,
                           const float* __restrict__ e_rowsum,
                           float* __restrict__ out, int N) {
    int wave = threadIdx.x >> 5;
    int lane = threadIdx.x & 31;
    int n = blockIdx.x * (blockDim.x >> 5) + wave;
    if (n >= N) return;

    float2 hp = *((const float2*)(h_prime + (size_t)n * 64) + lane);
    float rs = e_rowsum[n];
    float ux = hp.x / rs, uy = hp.y / rs;
    ux = (ux > 0.0f) ? ux : expm1f(ux);   // elu
    uy = (uy > 0.0f) ? uy : expm1f(uy);

    float un2 = ux * ux + uy * uy;
#pragma unroll
    for (int o = 16; o >= 1; o >>= 1) un2 += __shfl_xor(un2, o, 32);

    float u_norm = fmaxf(sqrtf(un2), MIN_NORM);
    float fac = tanhf(SQRT_C * u_norm) / (SQRT_C * u_norm);   // expmap0
    float px = fac * ux, py = fac * uy;
    float pn = fmaxf(fac * u_norm, MIN_NORM);                 // ||p|| (fac >= 0)
    float maxn = (1.0f - 4e-3f) / SQRT_C;                     // proj
    if (pn > maxn) {
        float s = maxn / pn;
        px *= s; py *= s;
    }
    float* o_ = out + (size_t)n * 64 + lane * 2;
    o_[0] = px;
    o_[1] = py;
}

// ---------------------------------------------------------------------------
extern "C" void kernel_launch(void* const* d_in, const int* in_sizes, int n_in,
                              void* d_out, int out_size, void* d_ws, size_t ws_size,
                              hipStream_t stream) {
    const float* x    = (const float*)d_in[0];
    const int*   edge = (const int*)d_in[1];
    const float* w    = (const float*)d_in[2];
    int N = in_sizes[0] / 64;
    int E = in_sizes[1] / 2;
    const int* erow = edge;
    const int* ecol = edge + E;

    // workspace layout (floats): h | h_e | edge_h | h_prime | e_rowsum | ssq
    float* h        = (float*)d_ws;
    float* h_e      = h + (size_t)N * 64;
    float* edge_h   = h_e + (size_t)N * 64;
    float* h_prime  = edge_h + (size_t)E;
    float* e_rowsum = h_prime + (size_t)N * 64;
    float* ssq      = e_rowsum + (size_t)N;

    int nz = N * 64 + N + 1;   // h_prime + e_rowsum + ssq, contiguous
    zero_kernel<<<(nz + 255) / 256, 256, 0, stream>>>(h_prime, nz);

    int ntiles = (N + 15) / 16;
    hyp_linear_kernel<<<(ntiles + 7) / 8, 256, 0, stream>>>(x, w, h, h_e, N, ntiles);

    edge_dist_kernel<<<(E + 7) / 8, 256, 0, stream>>>(h, erow, ecol, edge_h, E);

    ssq_kernel<<<1024, 256, 0, stream>>>(edge_h, ssq, E);

    scatter_kernel<<<(E + 7) / 8, 256, 0, stream>>>(edge_h, h_e, erow, ecol, ssq,
                                                    h_prime, e_rowsum, E);

    out_kernel<<<(N + 7) / 8, 256, 0, stream>>>(h_prime, e_rowsum, (float*)d_out, N);
}